// CantorAttention_58007828300069
// MI455X (gfx1250) — compile-verified
//
#include <hip/hip_runtime.h>
#include <hip/hip_bf16.h>

// ---------------------------------------------------------------------------
// CantorAttention for MI455X (gfx1250, wave32)
//   B=2, S=2048, DIM=512, H=8, HD=64, K=64
//   fp32 throughout; matrix work on V_WMMA_F32_16X16X4_F32
// ---------------------------------------------------------------------------

#define CA_B    2
#define CA_S    2048
#define CA_DIM  512
#define CA_H    8
#define CA_HD   64
#define CA_K    64
#define CA_M    (CA_B * CA_S)      // 4096 rows
#define CA_N1   (3 * CA_DIM)       // 1536 qkv cols
#define CA_SCALE 0.125f            // 1/sqrt(64)

typedef __attribute__((ext_vector_type(2))) float v2f;
typedef __attribute__((ext_vector_type(8))) float v8f;

// ---------------------------------------------------------------------------
// One wave computes a 16(M) x 64(N) fp32 tile of A(MxKd) * W(KdxN).
// A-fragment (16x4):  lanes 0-15 hold K=k0,k0+1 ; lanes 16-31 hold K=k0+2,k0+3
// B-fragment (4x16):  mirrored (N across lanes%16, K split across lane halves)
// C/D (16x16): VGPR r, lane -> row = r + 8*(lane>=16), col = lane%16
// ---------------------------------------------------------------------------
__device__ __forceinline__ void wmma_tile_16x64(const float* __restrict__ A,
                                                const float* __restrict__ W,
                                                int lda, int ldw,
                                                int m0, int n0, int Kd,
                                                v8f c[4]) {
  const int lane = threadIdx.x & 31;
  const int half = lane >> 4;   // 0 or 1
  const int l16  = lane & 15;
  const float* arow = A + (size_t)(m0 + l16) * lda;
  for (int k0 = 0; k0 < Kd; k0 += 4) {
    const int ka = k0 + half * 2;
    v2f a;
    a.x = arow[ka];
    a.y = arow[ka + 1];
    const float* wr0 = W + (size_t)ka * ldw;
    const float* wr1 = wr0 + ldw;
#pragma unroll
    for (int j = 0; j < 4; ++j) {
      const int ncol = n0 + j * 16 + l16;
      v2f b;
      b.x = wr0[ncol];
      b.y = wr1[ncol];
      // 8 args: (neg_a, A, neg_b, B, c_mod, C, reuse_a, reuse_b)
      c[j] = __builtin_amdgcn_wmma_f32_16x16x4_f32(
          false, a, false, b, (short)0, c[j], false, false);
    }
  }
}

// ---------------------------------------------------------------------------
// Kernel 1: qkv = x @ Wqkv + bqkv, scattered to (b,h,s,d) Q/K/V workspaces
// Grid: 256 m-tiles * 24 n64-tiles = 6144 wave-tiles; 4 waves/block
// ---------------------------------------------------------------------------
__global__ __launch_bounds__(128) void ca_qkv_gemm(
    const float* __restrict__ x, const float* __restrict__ Wqkv,
    const float* __restrict__ bqkv,
    float* __restrict__ qws, float* __restrict__ kws, float* __restrict__ vws) {
  const int wave   = blockIdx.x * (blockDim.x >> 5) + (threadIdx.x >> 5);
  const int tile_n = wave % (CA_N1 / 64);   // 0..23
  const int tile_m = wave / (CA_N1 / 64);   // 0..255
  const int m0 = tile_m * 16;
  const int n0 = tile_n * 64;

  v8f c[4] = {};
  wmma_tile_16x64(x, Wqkv, CA_DIM, CA_N1, m0, n0, CA_DIM, c);

  const int lane = threadIdx.x & 31;
  const int half = lane >> 4;
  const int l16  = lane & 15;
#pragma unroll
  for (int j = 0; j < 4; ++j) {
    const int N     = n0 + j * 16 + l16;
    const int which = N >> 9;          // 0=q 1=k 2=v
    const int h     = (N >> 6) & 7;
    const int d     = N & 63;
    float* dst = (which == 0) ? qws : (which == 1) ? kws : vws;
    const float bias = bqkv[N];
#pragma unroll
    for (int r = 0; r < 8; ++r) {
      const int M = m0 + r + half * 8;
      const int b = M >> 11;           // /2048
      const int s = M & 2047;
      dst[(((size_t)(b * CA_H + h) * CA_S) + s) * CA_HD + d] = c[j][r] + bias;
    }
  }
}

// ---------------------------------------------------------------------------
// Kernel 2: routed attention. One wave per query (b,h,s); 8 waves / block.
// ---------------------------------------------------------------------------
__global__ __launch_bounds__(256) void ca_attention(
    const float* __restrict__ qws, const float* __restrict__ kws,
    const float* __restrict__ vws, const int* __restrict__ routes,
    float* __restrict__ attn) {
  __shared__ float sh_q[8][CA_HD];
  __shared__ float sh_w[8][CA_K];
  __shared__ int   sh_r[8][CA_K];

  const int wid  = threadIdx.x >> 5;
  const int lane = threadIdx.x & 31;
  const int qid  = blockIdx.x * 8 + wid;   // 0..32767
  const int s    = qid & (CA_S - 1);
  const int bh   = qid >> 11;              // b*H + h

  const float* qp    = qws + ((size_t)bh * CA_S + s) * CA_HD;
  const float* kbase = kws + (size_t)bh * CA_S * CA_HD;
  const float* vbase = vws + (size_t)bh * CA_S * CA_HD;

  sh_q[wid][lane]      = qp[lane];
  sh_q[wid][lane + 32] = qp[lane + 32];
  sh_r[wid][lane]      = routes[s * CA_K + lane];
  sh_r[wid][lane + 32] = routes[s * CA_K + lane + 32];
  __syncthreads();

  // ---- scores: 2 keys per lane -------------------------------------------
  float sc[2];
#pragma unroll
  for (int t = 0; t < 2; ++t) {
    const int j = lane * 2 + t;
    const float* kp = kbase + (size_t)sh_r[wid][j] * CA_HD;
    float acc = 0.0f;
#pragma unroll 8
    for (int d = 0; d < CA_HD; ++d) acc += sh_q[wid][d] * kp[d];
    sc[t] = acc * CA_SCALE;
  }

  // ---- softmax over 64 (wave32 shuffle reduction, width 32!) -------------
  float mx = fmaxf(sc[0], sc[1]);
#pragma unroll
  for (int off = 16; off > 0; off >>= 1)
    mx = fmaxf(mx, __shfl_xor(mx, off, 32));
  const float e0 = __expf(sc[0] - mx);
  const float e1 = __expf(sc[1] - mx);
  float sum = e0 + e1;
#pragma unroll
  for (int off = 16; off > 0; off >>= 1)
    sum += __shfl_xor(sum, off, 32);
  const float inv = 1.0f / sum;
  sh_w[wid][lane * 2]     = e0 * inv;
  sh_w[wid][lane * 2 + 1] = e1 * inv;
  __syncthreads();

  // ---- out = w . V_gathered : 2 dims per lane ----------------------------
  const int b = bh >> 3, h = bh & 7;
#pragma unroll
  for (int t = 0; t < 2; ++t) {
    const int d = lane + 32 * t;
    float acc = 0.0f;
#pragma unroll 8
    for (int j = 0; j < CA_K; ++j)
      acc += sh_w[wid][j] * vbase[(size_t)sh_r[wid][j] * CA_HD + d];
    attn[((size_t)(b * CA_S + s)) * CA_DIM + h * CA_HD + d] = acc;
  }
}

// ---------------------------------------------------------------------------
// Kernel 3: out = attn @ Wout + bout  (M=4096, N=512, K=512)
// ---------------------------------------------------------------------------
__global__ __launch_bounds__(128) void ca_out_gemm(
    const float* __restrict__ attn, const float* __restrict__ Wout,
    const float* __restrict__ bout, float* __restrict__ out) {
  const int wave   = blockIdx.x * (blockDim.x >> 5) + (threadIdx.x >> 5);
  const int tile_n = wave & 7;             // 512/64 = 8 n-tiles
  const int tile_m = wave >> 3;            // 0..255
  const int m0 = tile_m * 16;
  const int n0 = tile_n * 64;

  v8f c[4] = {};
  wmma_tile_16x64(attn, Wout, CA_DIM, CA_DIM, m0, n0, CA_DIM, c);

  const int lane = threadIdx.x & 31;
  const int half = lane >> 4;
  const int l16  = lane & 15;
#pragma unroll
  for (int j = 0; j < 4; ++j) {
    const int N = n0 + j * 16 + l16;
    const float bias = bout[N];
#pragma unroll
    for (int r = 0; r < 8; ++r) {
      const int M = m0 + r + half * 8;
      out[(size_t)M * CA_DIM + N] = c[j][r] + bias;
    }
  }
}

// ---------------------------------------------------------------------------
extern "C" void kernel_launch(void* const* d_in, const int* in_sizes, int n_in,
                              void* d_out, int out_size, void* d_ws, size_t ws_size,
                              hipStream_t stream) {
  const float* x     = (const float*)d_in[0];
  const float* Wqkv  = (const float*)d_in[1];
  const float* bqkv  = (const float*)d_in[2];
  const float* Wout  = (const float*)d_in[3];
  const float* bout  = (const float*)d_in[4];
  const int*   routes= (const int*)d_in[5];

  const size_t head_elems = (size_t)CA_B * CA_H * CA_S * CA_HD;   // 2,097,152
  float* qws  = (float*)d_ws;
  float* kws  = qws + head_elems;
  float* vws  = kws + head_elems;
  float* attn = vws + head_elems;

  // GEMM1: 256 m-tiles * 24 n64-tiles = 6144 wave-tiles, 4 waves/block
  ca_qkv_gemm<<<6144 / 4, 128, 0, stream>>>(x, Wqkv, bqkv, qws, kws, vws);

  // Attention: 32768 queries, 8 waves/block
  ca_attention<<<(CA_B * CA_H * CA_S) / 8, 256, 0, stream>>>(qws, kws, vws,
                                                             routes, attn);

  // GEMM2: 256 m-tiles * 8 n64-tiles = 2048 wave-tiles, 4 waves/block
  ca_out_gemm<<<2048 / 4, 128, 0, stream>>>(attn, Wout, bout, (float*)d_out);
}